// RelationalKenn_16217796510109
// MI455X (gfx1250) — compile-verified
//
#include <hip/hip_runtime.h>
#include <math.h>
#include <stdint.h>

#define N_NODES 200000
#define N_EDGES 1000000
#define U 64

// ---------------------------------------------------------------------------
// CDNA5 async global<->LDS DMA via inline asm (ASYNCcnt-tracked, b128).
// VDST/VSRC VGPR carries the LDS *byte offset* (hardware adds the LDS base and the instruction offset).
// `tile` is the sole __shared__ allocation in the
// kernel, so its LDS base offset is 0 and &tile[i] <=> byte offset i*4.
// ---------------------------------------------------------------------------
#if defined(__gfx1250__)
#define HAVE_ASYNC_LDS 1
#else
#define HAVE_ASYNC_LDS 0
#endif

__device__ __forceinline__ void wait_async0() {
#if __has_builtin(__builtin_amdgcn_s_wait_asynccnt)
  __builtin_amdgcn_s_wait_asynccnt(0);
#else
  asm volatile("s_wait_asynccnt 0" ::: "memory");
#endif
}

#if HAVE_ASYNC_LDS
__device__ __forceinline__ void async_g2l_b128(uint32_t lds_byte_off, const void* g) {
  asm volatile("global_load_async_to_lds_b128 %0, %1, off"
               :
               : "v"(lds_byte_off), "v"((unsigned long long)(uintptr_t)g)
               : "memory");
}
__device__ __forceinline__ void async_l2g_b128(void* g, uint32_t lds_byte_off) {
  asm volatile("global_store_async_from_lds_b128 %0, %1, off"
               :
               : "v"((unsigned long long)(uintptr_t)g), "v"(lds_byte_off)
               : "memory");
}
#endif

__device__ __forceinline__ float2 sm2(float l0, float l1) {
  float m = fmaxf(l0, l1);
  float e0 = __expf(l0 - m), e1 = __expf(l1 - m);
  float r = 1.0f / (e0 + e1);
  return make_float2(e0 * r, e1 * r);
}

__device__ __forceinline__ float3 sm3(float l0, float l1, float l2) {
  float m = fmaxf(fmaxf(l0, l1), l2);
  float e0 = __expf(l0 - m), e1 = __expf(l1 - m), e2 = __expf(l2 - m);
  float r = 1.0f / (e0 + e1 + e2);
  return make_float3(e0 * r, e1 * r, e2 * r);
}

__device__ __forceinline__ void fadd_atomic(float* p, float v) {
  __hip_atomic_fetch_add(p, v, __ATOMIC_RELAXED, __HIP_MEMORY_SCOPE_AGENT);
}

// ---------------------------------------------------------------------------
// Kernel A: u = unary + unary_enhancer(unary). 64-row x 64-col tiles streamed
// through LDS with async global<->LDS DMA. LDS rows padded to 68 floats
// (keeps 16B alignment for b128, cuts bank conflicts in the clause phase).
// ---------------------------------------------------------------------------
__global__ __launch_bounds__(256) void kenn_unary(const float* __restrict__ unary,
                                                  const float* __restrict__ w_unary,
                                                  float* __restrict__ out_u) {
  __shared__ float tile[64 * 68];
  const int tid = threadIdx.x;
  const long long row0 = (long long)blockIdx.x * 64;

#if HAVE_ASYNC_LDS
  #pragma unroll
  for (int j = 0; j < 4; ++j) {
    int k = tid + 256 * j;           // 1024 float4 chunks per 64x64 tile
    int r = k >> 4, c4 = k & 15;
    async_g2l_b128((uint32_t)((r * 68 + c4 * 4) * 4),
                   unary + (row0 + r) * U + c4 * 4);
  }
  wait_async0();
  __syncthreads();
#else
  #pragma unroll
  for (int j = 0; j < 4; ++j) {
    int k = tid + 256 * j;
    int r = k >> 4, c4 = k & 15;
    *(float4*)&tile[r * 68 + c4 * 4] = *(const float4*)(unary + (row0 + r) * U + c4 * 4);
  }
  __syncthreads();
#endif

  if (tid < 64) {
    float* x = &tile[tid * 68];
    const float w0 = w_unary[0], w1 = w_unary[1], w2 = w_unary[2], w3 = w_unary[3];
    // Clause column sets are disjoint (0,1|2,3|4,5,6|7,8) -> in-place exact.
    { float2 s = sm2(-x[0], x[1]);        x[0] -= s.x * w0; x[1] += s.y * w0; }
    { float2 s = sm2(-x[2], x[3]);        x[2] -= s.x * w1; x[3] += s.y * w1; }
    { float3 s = sm3(-x[4], x[5], x[6]);  x[4] -= s.x * w2; x[5] += s.y * w2; x[6] += s.z * w2; }
    { float2 s = sm2(-x[7], x[8]);        x[7] -= s.x * w3; x[8] += s.y * w3; }
  }
  __syncthreads();   // s_wait_dscnt + barrier: LDS writes done before DMA-out

#if HAVE_ASYNC_LDS
  #pragma unroll
  for (int j = 0; j < 4; ++j) {
    int k = tid + 256 * j;
    int r = k >> 4, c4 = k & 15;
    async_l2g_b128(out_u + (row0 + r) * U + c4 * 4,
                   (uint32_t)((r * 68 + c4 * 4) * 4));
  }
  wait_async0();
#else
  #pragma unroll
  for (int j = 0; j < 4; ++j) {
    int k = tid + 256 * j;
    int r = k >> 4, c4 = k & 15;
    *(float4*)(out_u + (row0 + r) * U + c4 * 4) = *(const float4*)&tile[r * 68 + c4 * 4];
  }
#endif
}

// ---------------------------------------------------------------------------
// Kernel B: per-edge binary clauses + scatter. Recomputes u[i][0..3] locally
// from pristine `unary` (cols 0..3 depend only on unary clauses 0,1), so the
// gather never races the atomic scatter into out_u.
// Joined layout: cols 0..63 = u[i1], 64..127 = u[i2], 128..131 = binary.
// Clauses: (0,64) (1,65) (128,66) (3,129,67), sign -1 on first literal(s).
// ---------------------------------------------------------------------------
__global__ __launch_bounds__(256) void kenn_edges(const float* __restrict__ unary,
                                                  const float* __restrict__ binary,
                                                  const int* __restrict__ index1,
                                                  const int* __restrict__ index2,
                                                  const float* __restrict__ w_unary,
                                                  const float* __restrict__ w_binary,
                                                  float* __restrict__ out_u,
                                                  float* __restrict__ out_b) {
  const int e = blockIdx.x * blockDim.x + threadIdx.x;
  if (e >= N_EDGES) return;

  const int i1 = index1[e];
  const int i2 = index2[e];

  float4 a = *(const float4*)(unary + (long long)i1 * U);  // unary[i1][0..3]
  float4 b = *(const float4*)(unary + (long long)i2 * U);  // unary[i2][0..3]
  const float4 c = *(const float4*)(binary + (long long)e * 4);

  const float wu0 = w_unary[0], wu1 = w_unary[1];
  // Apply unary clauses 0 and 1 -> u[i][0..3]
  { float2 s = sm2(-a.x, a.y); a.x -= s.x * wu0; a.y += s.y * wu0; }
  { float2 s = sm2(-a.z, a.w); a.z -= s.x * wu1; a.w += s.y * wu1; }
  { float2 s = sm2(-b.x, b.y); b.x -= s.x * wu0; b.y += s.y * wu0; }
  { float2 s = sm2(-b.z, b.w); b.z -= s.x * wu1; b.w += s.y * wu1; }

  const float w0 = w_binary[0], w1 = w_binary[1], w2 = w_binary[2], w3 = w_binary[3];

  float ux0, ux1, ux3, uy0, uy1, uy2, uy3, db0, db1;
  { float2 s = sm2(-a.x, b.x);        ux0 = -s.x * w0; uy0 = s.y * w0; }
  { float2 s = sm2(-a.y, b.y);        ux1 = -s.x * w1; uy1 = s.y * w1; }
  { float2 s = sm2(-c.x, b.z);        db0 = -s.x * w2; uy2 = s.y * w2; }
  { float3 s = sm3(-a.w, -c.y, b.w);  ux3 = -s.x * w3; db1 = -s.y * w3; uy3 = s.z * w3; }

  float* p1 = out_u + (long long)i1 * U;
  float* p2 = out_u + (long long)i2 * U;
  fadd_atomic(p1 + 0, ux0);
  fadd_atomic(p1 + 1, ux1);
  fadd_atomic(p1 + 3, ux3);
  fadd_atomic(p2 + 0, uy0);
  fadd_atomic(p2 + 1, uy1);
  fadd_atomic(p2 + 2, uy2);
  fadd_atomic(p2 + 3, uy3);

  *(float4*)(out_b + (long long)e * 4) = make_float4(c.x + db0, c.y + db1, c.z, c.w);
}

extern "C" void kernel_launch(void* const* d_in, const int* in_sizes, int n_in,
                              void* d_out, int out_size, void* d_ws, size_t ws_size,
                              hipStream_t stream) {
  (void)in_sizes; (void)n_in; (void)out_size; (void)d_ws; (void)ws_size;
  const float* unary    = (const float*)d_in[0];
  const float* binary   = (const float*)d_in[1];
  const int*   index1   = (const int*)d_in[2];
  const int*   index2   = (const int*)d_in[3];
  const float* w_unary  = (const float*)d_in[4];
  const float* w_binary = (const float*)d_in[5];

  float* out_u = (float*)d_out;
  float* out_b = out_u + (size_t)N_NODES * U;

  // Kernel A fully overwrites out_u each call (resets atomic accumulation
  // between graph replays); kernel B fully overwrites out_b.
  hipLaunchKernelGGL(kenn_unary, dim3(N_NODES / 64), dim3(256), 0, stream,
                     unary, w_unary, out_u);
  hipLaunchKernelGGL(kenn_edges, dim3((N_EDGES + 255) / 256), dim3(256), 0, stream,
                     unary, binary, index1, index2, w_unary, w_binary, out_u, out_b);
}